// HybridMemory_57999238365647
// MI455X (gfx1250) — compile-verified
//
#include <hip/hip_runtime.h>
#include <math.h>

#define B_ROWS 1024
#define N_FEAT 131072
#define D_DIM  256
#define NC     8192
#define TEMP_F 0.05f
#define CSPLIT 8            // c-range splits -> gridDim.y, deterministic partial rowsums
#define CPB    64           // clusters owned per block in the scatter-free scan

typedef __attribute__((ext_vector_type(2))) float v2f;
typedef __attribute__((ext_vector_type(8))) float v8f;

__device__ __forceinline__ float waveReduceSum32(float v) {
    v += __shfl_xor(v, 16, 32);
    v += __shfl_xor(v,  8, 32);
    v += __shfl_xor(v,  4, 32);
    v += __shfl_xor(v,  2, 32);
    v += __shfl_xor(v,  1, 32);
    return v;
}

// ---------------- L2-normalize inputs: 1 wave per row ----------------
__global__ __launch_bounds__(256) void k_norm_inputs(const float* __restrict__ x,
                                                     float* __restrict__ out) {
    int wave = (blockIdx.x * blockDim.x + threadIdx.x) >> 5;  // 0..1023
    int lane = threadIdx.x & 31;
    const float* row = x + (size_t)wave * D_DIM + lane * 8;
    float f[8];
    #pragma unroll
    for (int j = 0; j < 8; ++j) f[j] = row[j];
    float ss = 0.0f;
    #pragma unroll
    for (int j = 0; j < 8; ++j) ss += f[j] * f[j];
    ss = waveReduceSum32(ss);
    float inv = 1.0f / fmaxf(sqrtf(ss), 1e-12f);
    float* o = out + (size_t)wave * D_DIM + lane * 8;
    #pragma unroll
    for (int j = 0; j < 8; ++j) o[j] = f[j] * inv;
}

// ------- owner-computes centroid scan: NO global atomics -------
// block b owns clusters [b*CPB, b*CPB+CPB); scans all labels; for each hit the
// whole wave normalizes the feature row and accumulates into LDS (ds_add_f32).
// Each feature row is read exactly once across the grid -> one 128MB HBM pass.
__global__ __launch_bounds__(256) void k_scan(const float* __restrict__ feat,
                                              const int* __restrict__ labels,
                                              float* __restrict__ cent,
                                              float* __restrict__ counts) {
    __shared__ float acc[CPB * D_DIM];   // 64KB
    __shared__ int   cnt[CPB];
    const int tid = threadIdx.x;
    const int cbase = blockIdx.x * CPB;

    for (int i = tid; i < CPB * D_DIM; i += 256) acc[i] = 0.0f;
    if (tid < CPB) cnt[tid] = 0;
    __syncthreads();

    const int w = tid >> 5, lane = tid & 31;
    // each wave handles label-chunks of 128 (int4 per lane), strided by 8 waves
    for (int it = w; it < N_FEAT / 128; it += 8) {
        const int sBase = it * 128;
        int4 lv = *(const int4*)(labels + sBase + lane * 4);
        int l4[4] = {lv.x, lv.y, lv.z, lv.w};
        #pragma unroll
        for (int j = 0; j < 4; ++j) {
            int rel = l4[j] - cbase;
            bool hit = ((unsigned)rel) < (unsigned)CPB;
            unsigned long long m = __ballot(hit);
            while (m) {                          // wave-uniform: m from ballot
                int src = __ffsll(m) - 1;
                m &= m - 1;
                int s = sBase + src * 4 + j;     // uniform row index
                int relc = __shfl(rel, src);
                // lane l owns columns {l, l+32, ...}: coalesced loads, 0-conflict LDS
                const float* row = feat + (size_t)s * D_DIM + lane;
                float f[8];
                #pragma unroll
                for (int q = 0; q < 8; ++q) f[q] = row[q * 32];
                float ss = 0.0f;
                #pragma unroll
                for (int q = 0; q < 8; ++q) ss += f[q] * f[q];
                ss = waveReduceSum32(ss);
                float inv = 1.0f / fmaxf(sqrtf(ss), 1e-12f);
                float* a = &acc[relc * D_DIM + lane];
                #pragma unroll
                for (int q = 0; q < 8; ++q) atomicAdd(a + q * 32, f[q] * inv);
                if (lane == 0) atomicAdd(&cnt[relc], 1);
            }
        }
    }
    __syncthreads();
    // plain stores: no k_zero pass needed anywhere
    for (int i = tid; i < CPB * D_DIM; i += 256)
        cent[(size_t)cbase * D_DIM + i] = acc[i];
    if (tid < CPB) counts[cbase + tid] = (float)cnt[tid];
}

// ---------------- per-cluster 1/(T*denom) and mask ----------------
__global__ void k_prep(const float* __restrict__ counts,
                       float* __restrict__ sinv, float* __restrict__ maskv) {
    int c = blockIdx.x * blockDim.x + threadIdx.x;
    if (c < NC) {
        float n = counts[c];
        float m = (n > 0.0f) ? 1.0f : 0.0f;
        float denom = m * n + (1.0f - m);
        sinv[c]  = 1.0f / (TEMP_F * denom);
        maskv[c] = m;
    }
}

// ------- WMMA fp32 GEMM over centroids + fused exp-row-sum (softmax denom) -------
// grid (64, CSPLIT): block = 16 input rows x (8192/CSPLIT) clusters; 8 waves;
// each wave runs 8 c-tiles with a 2-deep register-double-buffered B pipeline.
__global__ __launch_bounds__(256) void k_gemm(const float* __restrict__ inorm,
                                              const float* __restrict__ cent,
                                              const float* __restrict__ sinv,
                                              const float* __restrict__ maskv,
                                              float* __restrict__ rowsumPart) {
    __shared__ float shA[16 * 260];        // padded stride kills LDS bank conflicts
    __shared__ float partial[8][16];
    const int tid = threadIdx.x;
    const int b0 = blockIdx.x * 16;
    const int csBase = blockIdx.y * (NC / CSPLIT);   // 1024 clusters per split

    for (int i = tid; i < 16 * 256; i += 256) {
        int r = i >> 8, c = i & 255;
        shA[r * 260 + c] = inorm[(size_t)(b0 + r) * D_DIM + c];
    }
    __syncthreads();

    const int w = tid >> 5;
    const int lane = tid & 31;
    const int lm = lane & 15;      // lane's matrix row (A: M, B: N)
    const int lh = lane >> 4;      // K half-select: k-offset = 2*lh

    float acc[8];
    #pragma unroll
    for (int r = 0; r < 8; ++r) acc[r] = 0.0f;

    const float* aBase = &shA[lm * 260 + 2 * lh];
    // wave w owns tiles t=0..7: c0 = csBase + (w*8 + t)*16
    const float* bRow = cent + (size_t)(csBase + w * 8 * 16 + lm) * D_DIM + 2 * lh;

    // software pipeline: preload chunk 0 of tile 0 (8 x b64 -> 16 VGPRs/buffer)
    v2f bb[2][8];
    #pragma unroll
    for (int j = 0; j < 8; ++j) bb[0][j] = *(const v2f*)(bRow + j * 4);

    #pragma unroll
    for (int t = 0; t < 8; ++t) {
        const int c0 = csBase + (w * 8 + t) * 16;
        const float* bCur = bRow + (size_t)t * 16 * D_DIM;
        const float si = sinv[c0 + lm];      // issued early, used in epilogue
        const float mv = maskv[c0 + lm];
        v8f d = {};
        #pragma unroll
        for (int ch = 0; ch < 8; ++ch) {     // 8 chunks x 8 k-steps = K=256
            const int cur = ch & 1, nxt = cur ^ 1;
            if (ch < 7) {                    // prefetch next chunk of this tile
                const float* p = bCur + (ch + 1) * 32;
                #pragma unroll
                for (int j = 0; j < 8; ++j) bb[nxt][j] = *(const v2f*)(p + j * 4);
            } else if (t < 7) {              // prefetch chunk 0 of next tile
                const float* p = bCur + 16 * D_DIM;
                #pragma unroll
                for (int j = 0; j < 8; ++j) bb[nxt][j] = *(const v2f*)(p + j * 4);
            }
            #pragma unroll
            for (int j = 0; j < 8; ++j) {
                v2f a = *(const v2f*)(aBase + (ch * 8 + j) * 4);
                d = __builtin_amdgcn_wmma_f32_16x16x4_f32(
                        false, a, false, bb[cur][j], (short)0, d, false, false);
            }
        }
        #pragma unroll
        for (int r = 0; r < 8; ++r)
            acc[r] += expf(d[r] * si) * mv;  // exps = exp(sim)*mask
    }

    // reduce over the 16 clusters held across lanes of each half-wave
    #pragma unroll
    for (int r = 0; r < 8; ++r) {
        float v = acc[r];
        v += __shfl_xor(v, 1, 16);
        v += __shfl_xor(v, 2, 16);
        v += __shfl_xor(v, 4, 16);
        v += __shfl_xor(v, 8, 16);
        if (lane == 0)       partial[w][r]     = v;   // rows 0..7
        else if (lane == 16) partial[w][r + 8] = v;   // rows 8..15
    }
    __syncthreads();

    if (tid < 16) {
        float s = 0.0f;
        #pragma unroll
        for (int ww = 0; ww < 8; ++ww) s += partial[ww][tid];
        rowsumPart[(size_t)blockIdx.y * B_ROWS + b0 + tid] = s;  // deterministic partial
    }
}

// ------- per-sample loss: recompute target logit (one 256-dot per sample) -------
__global__ __launch_bounds__(256) void k_loss(const float* __restrict__ inorm,
                                              const float* __restrict__ cent,
                                              const float* __restrict__ sinv,
                                              const float* __restrict__ maskv,
                                              const float* __restrict__ rowsumPart,
                                              const int* __restrict__ indexes,
                                              const int* __restrict__ labels,
                                              float* __restrict__ lossb) {
    int wave = (blockIdx.x * blockDim.x + threadIdx.x) >> 5;  // 0..1023 = sample b
    int lane = threadIdx.x & 31;
    int tgt = labels[indexes[wave]];
    const float* xr = inorm + (size_t)wave * D_DIM + lane * 8;
    const float* cr = cent + (size_t)tgt * D_DIM + lane * 8;
    float dot = 0.0f;
    #pragma unroll
    for (int j = 0; j < 8; ++j) dot += xr[j] * cr[j];
    dot = waveReduceSum32(dot);
    if (lane == 0) {
        float rs = 0.0f;                    // fixed-order partial-sum reduce
        #pragma unroll
        for (int s = 0; s < CSPLIT; ++s) rs += rowsumPart[s * B_ROWS + wave];
        float t = dot * sinv[tgt];
        float e = expf(t) * maskv[tgt];
        float p = e / (rs + 1e-6f);
        lossb[wave] = -logf(p + 1e-6f);
    }
}

// ---------------- final mean over 1024 samples ----------------
__global__ void k_mean(const float* __restrict__ lossb, float* __restrict__ out) {
    __shared__ float sh[256];
    int t = threadIdx.x;
    float s = lossb[t] + lossb[t + 256] + lossb[t + 512] + lossb[t + 768];
    sh[t] = s;
    __syncthreads();
    for (int off = 128; off > 0; off >>= 1) {
        if (t < off) sh[t] += sh[t + off];
        __syncthreads();
    }
    if (t == 0) out[0] = sh[0] * (1.0f / 1024.0f);
}

extern "C" void kernel_launch(void* const* d_in, const int* in_sizes, int n_in,
                              void* d_out, int out_size, void* d_ws, size_t ws_size,
                              hipStream_t stream) {
    const float* inputs   = (const float*)d_in[0];   // [1024, 256]
    const float* features = (const float*)d_in[1];   // [131072, 256]
    const int*   indexes  = (const int*)d_in[2];     // [1024]
    const int*   labels   = (const int*)d_in[3];     // [131072]

    float* ws      = (float*)d_ws;
    float* inorm   = ws;                               // 1024*256
    float* cent    = inorm   + (size_t)B_ROWS * D_DIM; // 8192*256
    float* counts  = cent    + (size_t)NC * D_DIM;     // 8192
    float* sinv    = counts  + NC;                     // 8192
    float* maskv   = sinv    + NC;                     // 8192
    float* rowsumP = maskv   + NC;                     // CSPLIT*1024
    float* lossb   = rowsumP + CSPLIT * B_ROWS;        // 1024

    k_norm_inputs<<<B_ROWS / 8, 256, 0, stream>>>(inputs, inorm);
    k_scan<<<NC / CPB, 256, 0, stream>>>(features, labels, cent, counts);
    k_prep<<<NC / 256, 256, 0, stream>>>(counts, sinv, maskv);
    k_gemm<<<dim3(B_ROWS / 16, CSPLIT), 256, 0, stream>>>(inorm, cent, sinv, maskv, rowsumP);
    k_loss<<<B_ROWS / 8, 256, 0, stream>>>(inorm, cent, sinv, maskv, rowsumP,
                                           indexes, labels, lossb);
    k_mean<<<1, 256, 0, stream>>>(lossb, (float*)d_out);
}